// S2CNNFeature_21079699489120
// MI455X (gfx1250) — compile-verified
//
#include <hip/hip_runtime.h>
#include <cmath>
#include <vector>
#include <algorithm>

// ---------------------------------------------------------------------------
// S2CNN forward for MI455X (gfx1250, wave32, WMMA).
// Heavy GEMM-shaped work (batched 2D (I)DFTs of the SO(3) transforms and the
// per-l SO(3) convolution) runs on V_WMMA_F32_16X16X4_F32 with LDS-staged
// fragments. Wigner/quadrature tables are built on host per call (deterministic)
// and copied once. Scratch layout is hand-packed with region reuse (~230MB).
// ---------------------------------------------------------------------------

typedef float v2f __attribute__((ext_vector_type(2)));
typedef float v8f __attribute__((ext_vector_type(8)));

#define DEV __device__ __forceinline__

// D = A(16x4)*B(4x16) + C, full fp32. All modifier immediates must be 0 for f32.
DEV v8f wmma4p(v2f a, v2f b, v8f c) {
  return __builtin_amdgcn_wmma_f32_16x16x4_f32(false, a, false, b, (short)0, c,
                                               false, false);
}
// A-negated MAC (for the -Aim*Bim term of complex multiplies): f32 WMMA has no
// A/B NEG modifier, so negate the 2-element A fragment in VALU first.
DEV v8f wmma4n(v2f a, v2f b, v8f c) {
  v2f na;
  na.x = -a.x;
  na.y = -a.y;
  return __builtin_amdgcn_wmma_f32_16x16x4_f32(false, na, false, b, (short)0, c,
                                               false, false);
}

// A fragment: lane L -> A[mt*16 + (L&15)][kb + 2*(L>>4) + {0,1}]
DEV void frag_a(const float* A, int lda, int mt, int kb, int lane, v2f& o) {
  int r = mt * 16 + (lane & 15);
  int c = kb + 2 * (lane >> 4);
  o.x = A[r * lda + c];
  o.y = A[r * lda + c + 1];
}
// B fragment: lane L -> B[kb + 2*(L>>4) + {0,1}][nt*16 + (L&15)]
DEV void frag_b(const float* B, int ldb, int kb, int nt, int lane, v2f& o) {
  int r = kb + 2 * (lane >> 4);
  int c = nt * 16 + (lane & 15);
  o.x = B[r * ldb + c];
  o.y = B[(r + 1) * ldb + c];
}
// C store: VGPR j -> C[mt*16 + j + 8*(L>>4)][nt*16 + (L&15)]
DEV void store_c(float* C, int ldc, int mt, int nt, int lane, v8f c) {
  int r0 = mt * 16 + 8 * (lane >> 4), col = nt * 16 + (lane & 15);
#pragma unroll
  for (int j = 0; j < 8; j++) C[(r0 + j) * ldc + col] = c[j];
}

DEV int iabs_(int v) { return v < 0 ? -v : v; }
DEV int imax_(int a, int b) { return a > b ? a : b; }

// ---- table offsets (floats, relative to ws base) --------------------------
constexpr int T_DM0W = 0;         // 15360 : d^l_{m,0}(beta_in)*w_in      [l:60*(2l+1)]
constexpr int T_DEQM0 = 15360;    // 256   : d^l_{n,0}(pi/2)             [l*l + ni]
constexpr int T_DMFOLD = 15616;   // 174592: (2l+1)*d^l(beta_mid)        [p][5456]
constexpr int T_DMIDW = 190208;   // 42560 : d^l(beta_mid)*w_mid (l<10)  [p][1330]
constexpr int T_DEQ3 = 232768;    // 1330  : d^l(pi/2) full (l<10)
constexpr int T_DOUTF = 234098;   // 26600 : (2l+1)*d^l(beta_out)        [p][1330]
constexpr int T_WOUT = 260698;    // 20
constexpr int TAB_N = 260718;

// ---- workspace offsets (floats). planes: re at base, im at base+count -----
constexpr size_t O_IMG = 262144;    // 28800
constexpr size_t O_XC = 291000;     // 2*14880
constexpr size_t O_XHAT = 321000;   // 2*2048
constexpr size_t O_WFS2 = 325100;   // 2*3100
constexpr size_t O_BN1 = 331400;    // 200
constexpr size_t O_BN2 = 331600;    // 200
constexpr size_t O_XHAT2 = 332000;  // 2*1064000
constexpr size_t O_ZHAT2 = 2460000; // 2*1064000
constexpr size_t O_WFT = 4588000;   // 2*190000
constexpr size_t BIG = 5000000;
constexpr size_t O_ZHAT = BIG;              // 2*4364800 (dead after so3_ifft1)
constexpr size_t O_Z = BIG + 9000000;       // 26214400  (dead after so3_fft)
constexpr size_t O_U2 = BIG + 36000000;     // 2*9241600 (dead after wigf)
constexpr size_t O_AT = BIG;                // 2*288800  (reuses ZHAT region)
constexpr size_t O_G = BIG + 600000;        // 2*5487200 (reuses Z region)
constexpr size_t O_Z2 = BIG + 13000000;     // 6400000
// peak: (BIG + 54,483,200) * 4B  ~= 228 MB of d_ws

constexpr float PI2 = 6.28318530717958647692f;

// ===========================================================================
// small kernels
// ===========================================================================
__global__ void k_zero(float* p, int n) {
  int i = blockIdx.x * 256 + threadIdx.x;
  if (i < n) p[i] = 0.f;
}

__global__ void k_project(const float* __restrict__ x, float* __restrict__ img) {
  int i = blockIdx.x * 256 + threadIdx.x;
  if (i >= 8 * 2048) return;
  int b = i >> 11, n = i & 2047;
  float xv = x[(b * 3 + 0) * 2048 + n];
  float yv = x[(b * 3 + 1) * 2048 + n];
  float zv = x[(b * 3 + 2) * 2048 + n];
  float r = sqrtf(xv * xv + yv * yv + zv * zv + 1e-12f);
  float cz = zv / r;
  cz = fminf(fmaxf(cz, -1.f + 1e-7f), 1.f - 1e-7f);
  float beta = acosf(cz);
  float alpha = atan2f(yv, xv);
  if (alpha < 0.f) alpha += PI2;
  int jb = (int)(beta * (60.f / 3.14159265358979323846f));
  jb = jb < 0 ? 0 : (jb > 59 ? 59 : jb);
  int ja = (int)(alpha * (60.f / PI2));
  ja = ja < 0 ? 0 : (ja > 59 ? 59 : ja);
  // r > 0 and img init 0: uint bit-pattern order == float order
  atomicMax((unsigned int*)&img[(b * 60 + jb) * 60 + ja], __float_as_uint(r));
}

// alpha-DFT of projected image: Xc[b,p,m] m in [-15,15]
__global__ void k_adft(float* __restrict__ W) {
  int i = blockIdx.x * 256 + threadIdx.x;
  if (i >= 14880) return;
  int bp = i / 31, mb = i - bp * 31;
  int m = mb - 15;
  const float* img = W + O_IMG + bp * 60;
  float sr = 0.f, si = 0.f;
  for (int a = 0; a < 60; a++) {
    float v = img[a];
    float ang = -(PI2 / 60.f) * (float)(m * a);
    float s, c;
    sincosf(ang, &s, &c);
    sr += v * c;
    si += v * s;
  }
  W[O_XC + i] = sr;
  W[O_XC + 14880 + i] = si;
}

// Legendre contraction: xhat[b, l*l+mi]
__global__ void k_leg(float* __restrict__ W) {
  int i = blockIdx.x * 256 + threadIdx.x;
  if (i >= 2048) return;
  int b = i >> 8, t = i & 255;
  int l = 0;
  while ((l + 1) * (l + 1) <= t) l++;
  int mi = t - l * l, w = 2 * l + 1, m = mi - l;
  const float* dw = W + T_DM0W + 60 * l * l;
  float sr = 0.f, si = 0.f;
  for (int p = 0; p < 60; p++) {
    float d = dw[p * w + mi];
    int xi = (b * 60 + p) * 31 + (m + 15);
    sr += d * W[O_XC + xi];
    si += d * W[O_XC + 14880 + xi];
  }
  W[O_XHAT + b * 256 + t] = sr;
  W[O_XHAT + 2048 + b * 256 + t] = si;
}

// alpha-DFT of scaled s2 filters: wfs2[m+15, o]
__global__ void k_wfs2(const float* __restrict__ w_s2, float* __restrict__ W) {
  int i = blockIdx.x * 256 + threadIdx.x;
  if (i >= 3100) return;
  int nb = i / 100, o = i - nb * 100;
  int m = nb - 15;
  float sc = rsqrtf(60.0f * 65536.0f / 900.0f);  // 1/sqrt(k_s2*B_MID^4/B_IN^2)
  float sr = 0.f, si = 0.f;
  for (int k = 0; k < 60; k++) {
    float v = w_s2[o * 60 + k] * sc;
    float ang = -(PI2 / 60.f) * (float)(k * m);
    float s, c;
    sincosf(ang, &s, &c);
    sr += v * c;
    si += v * s;
  }
  W[O_WFS2 + i] = sr;
  W[O_WFS2 + 3100 + i] = si;
}

// S2 conv (F_in = 1): zhat[bo, packed(l,mi,ni)] = xhat[b,l,mi] * conj(deq*wfs2)
__global__ void k_s2conv(float* __restrict__ W) {
  int i = blockIdx.x * 256 + threadIdx.x;
  if (i >= 4364800) return;
  int bo = i / 5456, t = i - bo * 5456;
  int l = 0;
  while ((l + 1) * (4 * (l + 1) * (l + 1) - 1) / 3 <= t) l++;
  int off = l * (4 * l * l - 1) / 3, w = 2 * l + 1;
  int rem = t - off, mi = rem / w, ni = rem - mi * w;
  int b = bo / 100, o = bo - b * 100;
  float xr = W[O_XHAT + b * 256 + l * l + mi];
  float xi = W[O_XHAT + 2048 + b * 256 + l * l + mi];
  float dq = W[T_DEQM0 + l * l + ni];
  int wi = (ni - l + 15) * 100 + o;
  float gr = dq * W[O_WFS2 + wi];
  float gi = dq * W[O_WFS2 + 3100 + wi];
  W[O_ZHAT + (size_t)bo * 5456 + t] = xr * gr + xi * gi;
  W[O_ZHAT + 4364800 + (size_t)bo * 5456 + t] = xi * gr - xr * gi;
}

// ===========================================================================
// fused SO(3) inverse FFT, b=16: per (bf, p) block: Wigner l-sum -> E(31x31) in
// LDS, then z[32x32] = Re( IDFT_a * E * IDFT_g ) via f32 WMMA. + bias.
// ===========================================================================
__global__ void __launch_bounds__(128) k_so3_ifft1(float* __restrict__ W,
                                                   const float* __restrict__ b_s2) {
  __shared__ float ls[8 * 1024];
  float *ERe = ls, *EIm = ls + 1024, *WaRe = ls + 2048, *WaIm = ls + 3072,
        *WgRe = ls + 4096, *WgIm = ls + 5120, *TRe = ls + 6144, *TIm = ls + 7168;
  int bf = blockIdx.x >> 5, p = blockIdx.x & 31;
  int tid = threadIdx.x, lane = tid & 31, wv = tid >> 5;
  const float* zr = W + O_ZHAT + (size_t)bf * 5456;
  const float* zi = zr + 4364800;
  const float* df = W + T_DMFOLD + p * 5456;
  for (int idx = tid; idx < 1024; idx += 128) {
    int r = idx >> 5, c = idx & 31;
    float er = 0.f, ei = 0.f;
    if (r < 31 && c < 31) {
      int m = r - 15, n = c - 15;
      int lmin = imax_(iabs_(m), iabs_(n));
      for (int l = lmin; l < 16; l++) {
        int w = 2 * l + 1;
        int j = l * (4 * l * l - 1) / 3 + (m + l) * w + (n + l);
        float d = df[j];
        er += d * zr[j];
        ei += d * zi[j];
      }
    }
    ERe[idx] = er;
    EIm[idx] = ei;
    float wr = 0.f, wi2 = 0.f;
    if (c < 31) {
      float ang = (PI2 / 32.f) * (float)(r * (c - 15));
      sincosf(ang, &wi2, &wr);
    }
    WaRe[idx] = wr;
    WaIm[idx] = wi2;
    float gr = 0.f, gi = 0.f;
    if (r < 31) {
      float ang = (PI2 / 32.f) * (float)(c * (r - 15));
      sincosf(ang, &gi, &gr);
    }
    WgRe[idx] = gr;
    WgIm[idx] = gi;
  }
  __syncthreads();
  int Mt = wv >> 1, Nt = wv & 1;
  v8f cre = {}, cim = {};
  for (int kk = 0; kk < 8; kk++) {
    int kb = kk * 4;
    v2f ar, ai, br, bi;
    frag_a(WaRe, 32, Mt, kb, lane, ar);
    frag_a(WaIm, 32, Mt, kb, lane, ai);
    frag_b(ERe, 32, kb, Nt, lane, br);
    frag_b(EIm, 32, kb, Nt, lane, bi);
    cre = wmma4p(ar, br, cre);
    cre = wmma4n(ai, bi, cre);
    cim = wmma4p(ar, bi, cim);
    cim = wmma4p(ai, br, cim);
  }
  store_c(TRe, 32, Mt, Nt, lane, cre);
  store_c(TIm, 32, Mt, Nt, lane, cim);
  __syncthreads();
  v8f cz = {};
  for (int kk = 0; kk < 8; kk++) {
    int kb = kk * 4;
    v2f ar, ai, br, bi;
    frag_a(TRe, 32, Mt, kb, lane, ar);
    frag_a(TIm, 32, Mt, kb, lane, ai);
    frag_b(WgRe, 32, kb, Nt, lane, br);
    frag_b(WgIm, 32, kb, Nt, lane, bi);
    cz = wmma4p(ar, br, cz);
    cz = wmma4n(ai, bi, cz);
  }
  int o = bf - (bf / 100) * 100;
  float bias = b_s2[o];
  float* zout = W + O_Z + (size_t)bf * 32768 + p * 1024;
  int row0 = Mt * 16 + 8 * (lane >> 4), col = Nt * 16 + (lane & 15);
#pragma unroll
  for (int j = 0; j < 8; j++)
    zout[(row0 + j) * 32 + col] = cz[j] * (1.f / 1024.f) + bias;
}

// ===========================================================================
// BN stats / apply (+ReLU)
// ===========================================================================
__global__ void k_bn_stats(const float* __restrict__ z, float* __restrict__ st, int S) {
  int o = blockIdx.x;
  double s = 0.0, s2 = 0.0;
  for (int idx = threadIdx.x; idx < 8 * S; idx += 256) {
    int b = idx / S, r = idx - b * S;
    double v = (double)z[(size_t)(b * 100 + o) * S + r];
    s += v;
    s2 += v * v;
  }
  __shared__ double sh[512];
  sh[threadIdx.x] = s;
  sh[256 + threadIdx.x] = s2;
  __syncthreads();
  for (int off = 128; off; off >>= 1) {
    if (threadIdx.x < off) {
      sh[threadIdx.x] += sh[threadIdx.x + off];
      sh[256 + threadIdx.x] += sh[256 + threadIdx.x + off];
    }
    __syncthreads();
  }
  if (!threadIdx.x) {
    double N = 8.0 * S;
    double mu = sh[0] / N;
    double var = sh[256] / N - mu * mu;
    st[o] = (float)mu;
    st[100 + o] = (float)var;
  }
}

__global__ void k_bn_apply(float* __restrict__ z, const float* __restrict__ st,
                           const float* __restrict__ g, const float* __restrict__ be,
                           int S, int total) {
  int i = blockIdx.x * 256 + threadIdx.x;
  if (i >= total) return;
  int bf = i / S;
  int o = bf - (bf / 100) * 100;
  float xn = (z[i] - st[o]) * rsqrtf(st[100 + o] + 1e-5f);
  float y = xn * g[o] + be[o];
  z[i] = fmaxf(y, 0.f);
}

// ===========================================================================
// fused forward SO(3) FFT, b=16: per (bf,p): U2[19x19] = F2a * (z * Fg), WMMA.
// ===========================================================================
__global__ void __launch_bounds__(128) k_so3_fft(float* __restrict__ W) {
  __shared__ float ls[7 * 1024];
  float *Zl = ls, *FgRe = ls + 1024, *FgIm = ls + 2048, *U1Re = ls + 3072,
        *U1Im = ls + 4096, *F2Re = ls + 5120, *F2Im = ls + 6144;
  int bf = blockIdx.x >> 5, p = blockIdx.x & 31;
  int tid = threadIdx.x, lane = tid & 31, wv = tid >> 5;
  const float* zin = W + O_Z + (size_t)bf * 32768 + p * 1024;
  for (int idx = tid; idx < 1024; idx += 128) {
    Zl[idx] = zin[idx];
    int r = idx >> 5, c = idx & 31;
    float fr = 0.f, fi = 0.f;
    if (c < 19) {
      float ang = -(PI2 / 32.f) * (float)(r * (c - 9));
      sincosf(ang, &fi, &fr);
    }
    FgRe[idx] = fr;
    FgIm[idx] = fi;
    float gr = 0.f, gi = 0.f;
    if (r < 19) {
      float ang = -(PI2 / 32.f) * (float)((r - 9) * c);
      sincosf(ang, &gi, &gr);
    }
    F2Re[idx] = gr;
    F2Im[idx] = gi;
  }
  __syncthreads();
  int Mt = wv >> 1, Nt = wv & 1;
  v8f ur = {}, ui = {};
  for (int kk = 0; kk < 8; kk++) {
    int kb = kk * 4;
    v2f az, br, bi;
    frag_a(Zl, 32, Mt, kb, lane, az);
    frag_b(FgRe, 32, kb, Nt, lane, br);
    frag_b(FgIm, 32, kb, Nt, lane, bi);
    ur = wmma4p(az, br, ur);
    ui = wmma4p(az, bi, ui);
  }
  store_c(U1Re, 32, Mt, Nt, lane, ur);
  store_c(U1Im, 32, Mt, Nt, lane, ui);
  __syncthreads();
  v8f cre = {}, cim = {};
  for (int kk = 0; kk < 8; kk++) {
    int kb = kk * 4;
    v2f ar, ai, br, bi;
    frag_a(F2Re, 32, Mt, kb, lane, ar);
    frag_a(F2Im, 32, Mt, kb, lane, ai);
    frag_b(U1Re, 32, kb, Nt, lane, br);
    frag_b(U1Im, 32, kb, Nt, lane, bi);
    cre = wmma4p(ar, br, cre);
    cre = wmma4n(ai, bi, cre);
    cim = wmma4p(ar, bi, cim);
    cim = wmma4p(ai, br, cim);
  }
  int row0 = Mt * 16 + 8 * (lane >> 4), col = Nt * 16 + (lane & 15);
  float* u2r = W + O_U2 + (size_t)(bf * 32 + p) * 361;
#pragma unroll
  for (int j = 0; j < 8; j++) {
    int mr = row0 + j;
    if (mr < 19 && col < 19) {
      u2r[mr * 19 + col] = cre[j];
      u2r[9241600 + mr * 19 + col] = cim[j];
    }
  }
}

// Wigner forward: xhat2[bf, packed3(l,mi,ki)] = sum_p dmidw * U2
__global__ void k_wigf(float* __restrict__ W) {
  int i = blockIdx.x * 256 + threadIdx.x;
  if (i >= 1064000) return;
  int bf = i / 1330, t = i - bf * 1330;
  int l = 0;
  while ((l + 1) * (4 * (l + 1) * (l + 1) - 1) / 3 <= t) l++;
  int off = l * (4 * l * l - 1) / 3, w = 2 * l + 1;
  int rem = t - off, mi = rem / w, ki = rem - mi * w;
  int ub = (mi - l + 9) * 19 + (ki - l + 9);
  float sr = 0.f, si = 0.f;
  for (int p = 0; p < 32; p++) {
    float d = W[T_DMIDW + p * 1330 + t];
    size_t ui = O_U2 + (size_t)(bf * 32 + p) * 361 + ub;
    sr += d * W[ui];
    si += d * W[ui + 9241600];
  }
  W[O_XHAT2 + i] = sr;
  W[O_XHAT2 + 1064000 + i] = si;
}

// alpha-DFT of scaled SO(3) filters, stored conjugated: WfT[i][n+9,o]
__global__ void k_wft(const float* __restrict__ w_so3, float* __restrict__ W) {
  int i = blockIdx.x * 256 + threadIdx.x;
  if (i >= 190000) return;
  int ii = i / 1900, rest = i - ii * 1900;
  int nb = rest / 100, o = rest - nb * 100;
  int m = nb - 9;
  float sc = rsqrtf(32.0f * 100.0f * 1000.0f / 4096.0f);  // 1/sqrt(k*F1*BO^3/BM^3)
  float sr = 0.f, si = 0.f;
  for (int k = 0; k < 32; k++) {
    float v = w_so3[(ii * 100 + o) * 32 + k] * sc;
    float ang = -(PI2 / 32.f) * (float)(k * m);
    float s, c;
    sincosf(ang, &s, &c);
    sr += v * c;
    si += v * s;
  }
  W[O_WFT + i] = sr;
  W[O_WFT + 190000 + i] = -si;  // conj
}

// transpose xhat2 into GEMM A layout: At[(b,mi,ki)][i]
__global__ void k_at(float* __restrict__ W, int l) {
  int w = 2 * l + 1, M = 8 * w * w;
  int i = blockIdx.x * 256 + threadIdx.x;
  if (i >= M * 100) return;
  int r = i / 100, ii = i - r * 100;
  int b = r / (w * w);
  int rem = r - b * (w * w);
  int mi = rem / w, ki = rem - mi * w;
  int off3 = l * (4 * l * l - 1) / 3;
  size_t src = O_XHAT2 + (size_t)(b * 100 + ii) * 1330 + off3 + mi * w + ki;
  W[O_AT + i] = W[src];
  W[O_AT + 288800 + i] = W[src + 1064000];
}

// generic complex GEMM C = A*B via f32 WMMA; 32x64 block tile, K panels of 16.
__global__ void __launch_bounds__(256) k_cgemm(
    const float* __restrict__ Ar, const float* __restrict__ Ai,
    const float* __restrict__ Br, const float* __restrict__ Bi,
    float* __restrict__ Cr, float* __restrict__ Ci, int M, int N, int K,
    int lda, int ldb, int ldc, int bcol) {
  __shared__ float sm[3072];
  float *Asr = sm, *Asi = sm + 512, *Bsr = sm + 1024, *Bsi = sm + 2048;
  int tid = threadIdx.x, lane = tid & 31, wv = tid >> 5;
  int m0 = blockIdx.x * 32, n0 = blockIdx.y * 64;
  int Mt = wv >> 2, Nt = wv & 3;
  v8f cre = {}, cim = {};
  int KT = (K + 15) >> 4;
  for (int kt = 0; kt < KT; kt++) {
    int k0 = kt * 16;
    for (int idx = tid; idx < 512; idx += 256) {
      int r = idx >> 4, c = idx & 15;
      int gm = m0 + r, k = k0 + c;
      bool ok = (gm < M) && (k < K);
      Asr[idx] = ok ? Ar[(size_t)gm * lda + k] : 0.f;
      Asi[idx] = ok ? Ai[(size_t)gm * lda + k] : 0.f;
    }
    for (int idx = tid; idx < 1024; idx += 256) {
      int r = idx >> 6, c = idx & 63;
      int k = k0 + r, gn = n0 + c;
      bool ok = (k < K) && (gn < N);
      Bsr[idx] = ok ? Br[(size_t)k * ldb + bcol + gn] : 0.f;
      Bsi[idx] = ok ? Bi[(size_t)k * ldb + bcol + gn] : 0.f;
    }
    __syncthreads();
    for (int kk = 0; kk < 4; kk++) {
      int kb = kk * 4;
      v2f ar, ai, br, bi;
      frag_a(Asr, 16, Mt, kb, lane, ar);
      frag_a(Asi, 16, Mt, kb, lane, ai);
      frag_b(Bsr, 64, kb, Nt, lane, br);
      frag_b(Bsi, 64, kb, Nt, lane, bi);
      cre = wmma4p(ar, br, cre);
      cre = wmma4n(ai, bi, cre);
      cim = wmma4p(ar, bi, cim);
      cim = wmma4p(ai, br, cim);
    }
    __syncthreads();
  }
  int row0 = m0 + Mt * 16 + 8 * (lane >> 4), col = n0 + Nt * 16 + (lane & 15);
#pragma unroll
  for (int j = 0; j < 8; j++) {
    int r = row0 + j;
    if (r < M && col < N) {
      Cr[(size_t)r * ldc + col] = cre[j];
      Ci[(size_t)r * ldc + col] = cim[j];
    }
  }
}

// k-contraction with d_eq(pi/2): zhat2[bo, packed3] = sum_k d[n,k]*G[(b,m,k),(n,o)]
__global__ void k_kred(float* __restrict__ W, int l) {
  int w = 2 * l + 1;
  int total = 800 * w * w;
  int i = blockIdx.x * 256 + threadIdx.x;
  if (i >= total) return;
  int bo = i / (w * w);
  int rem = i - bo * (w * w);
  int mi = rem / w, ni = rem - mi * w;
  int b = bo / 100, o = bo - b * 100;
  int off3 = l * (4 * l * l - 1) / 3;
  int Nl = 100 * w;
  float sr = 0.f, si = 0.f;
  for (int ki = 0; ki < w; ki++) {
    float d = W[T_DEQ3 + off3 + ni * w + ki];
    size_t gi = O_G + (size_t)((b * w + mi) * w + ki) * Nl + ni * 100 + o;
    sr += d * W[gi];
    si += d * W[gi + 5487200];
  }
  size_t dst = O_ZHAT2 + (size_t)bo * 1330 + off3 + mi * w + ni;
  W[dst] = sr;
  W[dst + 1064000] = si;
}

// ===========================================================================
// fused SO(3) inverse FFT, b=10 (same structure as ifft1, K padded to 20)
// ===========================================================================
__global__ void __launch_bounds__(128) k_so3_ifft2(float* __restrict__ W,
                                                   const float* __restrict__ b_so3) {
  __shared__ float ls[8 * 1024];
  float *ERe = ls, *EIm = ls + 1024, *WaRe = ls + 2048, *WaIm = ls + 3072,
        *WgRe = ls + 4096, *WgIm = ls + 5120, *TRe = ls + 6144, *TIm = ls + 7168;
  int bf = blockIdx.x / 20, p = blockIdx.x - bf * 20;
  int tid = threadIdx.x, lane = tid & 31, wv = tid >> 5;
  const float* zr = W + O_ZHAT2 + (size_t)bf * 1330;
  const float* zi = zr + 1064000;
  const float* df = W + T_DOUTF + p * 1330;
  for (int idx = tid; idx < 1024; idx += 128) {
    int r = idx >> 5, c = idx & 31;
    float er = 0.f, ei = 0.f;
    if (r < 19 && c < 19) {
      int m = r - 9, n = c - 9;
      int lmin = imax_(iabs_(m), iabs_(n));
      for (int l = lmin; l < 10; l++) {
        int w = 2 * l + 1;
        int j = l * (4 * l * l - 1) / 3 + (m + l) * w + (n + l);
        float d = df[j];
        er += d * zr[j];
        ei += d * zi[j];
      }
    }
    ERe[idx] = er;
    EIm[idx] = ei;
    float wr = 0.f, wi2 = 0.f;
    if (r < 20 && c < 19) {
      float ang = (PI2 / 20.f) * (float)(r * (c - 9));
      sincosf(ang, &wi2, &wr);
    }
    WaRe[idx] = wr;
    WaIm[idx] = wi2;
    float gr = 0.f, gi = 0.f;
    if (r < 19 && c < 20) {
      float ang = (PI2 / 20.f) * (float)(c * (r - 9));
      sincosf(ang, &gi, &gr);
    }
    WgRe[idx] = gr;
    WgIm[idx] = gi;
  }
  __syncthreads();
  int Mt = wv >> 1, Nt = wv & 1;
  v8f cre = {}, cim = {};
  for (int kk = 0; kk < 5; kk++) {  // K padded to 20
    int kb = kk * 4;
    v2f ar, ai, br, bi;
    frag_a(WaRe, 32, Mt, kb, lane, ar);
    frag_a(WaIm, 32, Mt, kb, lane, ai);
    frag_b(ERe, 32, kb, Nt, lane, br);
    frag_b(EIm, 32, kb, Nt, lane, bi);
    cre = wmma4p(ar, br, cre);
    cre = wmma4n(ai, bi, cre);
    cim = wmma4p(ar, bi, cim);
    cim = wmma4p(ai, br, cim);
  }
  store_c(TRe, 32, Mt, Nt, lane, cre);
  store_c(TIm, 32, Mt, Nt, lane, cim);
  __syncthreads();
  v8f cz = {};
  for (int kk = 0; kk < 5; kk++) {
    int kb = kk * 4;
    v2f ar, ai, br, bi;
    frag_a(TRe, 32, Mt, kb, lane, ar);
    frag_a(TIm, 32, Mt, kb, lane, ai);
    frag_b(WgRe, 32, kb, Nt, lane, br);
    frag_b(WgIm, 32, kb, Nt, lane, bi);
    cz = wmma4p(ar, br, cz);
    cz = wmma4n(ai, bi, cz);
  }
  int o = bf - (bf / 100) * 100;
  float bias = b_so3[o];
  int row0 = Mt * 16 + 8 * (lane >> 4), col = Nt * 16 + (lane & 15);
  float* zout = W + O_Z2 + (size_t)bf * 8000 + p * 400;
#pragma unroll
  for (int j = 0; j < 8; j++) {
    int a = row0 + j;
    if (a < 20 && col < 20) zout[a * 20 + col] = cz[j] * (1.f / 400.f) + bias;
  }
}

// final quadrature reduction: out[b,c] = sum_{p,a,g} z2 * w_out[p]
__global__ void k_final(const float* __restrict__ W, float* __restrict__ out) {
  int bc = blockIdx.x;
  const float* z2 = W + O_Z2 + (size_t)bc * 8000;
  float s = 0.f;
  for (int idx = threadIdx.x; idx < 8000; idx += 256)
    s += z2[idx] * W[T_WOUT + idx / 400];
  __shared__ float sh[256];
  sh[threadIdx.x] = s;
  __syncthreads();
  for (int off = 128; off; off >>= 1) {
    if (threadIdx.x < off) sh[threadIdx.x] += sh[threadIdx.x + off];
    __syncthreads();
  }
  if (!threadIdx.x) out[bc] = sh[0];
}

// ===========================================================================
// host: Wigner-d / quadrature tables (deterministic, rebuilt every call)
// ===========================================================================
static void build_tables(float* T) {
  double fact[36];
  fact[0] = 1.0;
  for (int i = 1; i < 36; i++) fact[i] = fact[i - 1] * i;
  auto wd = [&](int l, int mp, int m, double beta) -> double {
    double ch = cos(beta * 0.5), sh = sin(beta * 0.5);
    double pre = sqrt(fact[l + mp] * fact[l - mp] * fact[l + m] * fact[l - m]);
    int kmin = std::max(0, m - mp), kmax = std::min(l + m, l - mp);
    double sum = 0.0;
    for (int k = kmin; k <= kmax; k++) {
      double t = pow(ch, 2 * l + m - mp - 2 * k) * pow(sh, mp - m + 2 * k) /
                 (fact[l + m - k] * fact[k] * fact[mp - m + k] * fact[l - mp - k]);
      sum += ((mp - m + k) & 1) ? -t : t;
    }
    return pre * sum;
  };
  auto qwf = [&](int b, double* o) {
    for (int j = 0; j < 2 * b; j++) {
      double s = 0.0;
      for (int k = 0; k < b; k++)
        s += sin(M_PI * (2 * j + 1) * (2 * k + 1) / (4.0 * b)) / (2 * k + 1);
      o[j] = 2.0 / b * sin(M_PI * (2 * j + 1) / (4.0 * b)) * s;
    }
  };
  double win[60], wmid[32], wout[20], bin[60], bmid[32], bout[20];
  qwf(30, win);
  qwf(16, wmid);
  qwf(10, wout);
  for (int j = 0; j < 60; j++) bin[j] = (j + 0.5) * M_PI / 60.0;
  for (int j = 0; j < 32; j++) bmid[j] = (j + 0.5) * M_PI / 32.0;
  for (int j = 0; j < 20; j++) bout[j] = (j + 0.5) * M_PI / 20.0;
  for (int l = 0; l < 16; l++) {
    int w = 2 * l + 1;
    for (int p = 0; p < 60; p++)
      for (int mi = 0; mi < w; mi++)
        T[T_DM0W + 60 * l * l + p * w + mi] = (float)(wd(l, mi - l, 0, bin[p]) * win[p]);
    for (int ni = 0; ni < w; ni++)
      T[T_DEQM0 + l * l + ni] = (float)wd(l, ni - l, 0, M_PI * 0.5);
  }
  for (int p = 0; p < 32; p++)
    for (int l = 0; l < 16; l++) {
      int w = 2 * l + 1, off = l * (4 * l * l - 1) / 3;
      for (int mi = 0; mi < w; mi++)
        for (int ni = 0; ni < w; ni++)
          T[T_DMFOLD + p * 5456 + off + mi * w + ni] =
              (float)((2 * l + 1) * wd(l, mi - l, ni - l, bmid[p]));
    }
  for (int p = 0; p < 32; p++)
    for (int l = 0; l < 10; l++) {
      int w = 2 * l + 1, off = l * (4 * l * l - 1) / 3;
      for (int mi = 0; mi < w; mi++)
        for (int ki = 0; ki < w; ki++)
          T[T_DMIDW + p * 1330 + off + mi * w + ki] =
              (float)(wd(l, mi - l, ki - l, bmid[p]) * wmid[p]);
    }
  for (int l = 0; l < 10; l++) {
    int w = 2 * l + 1, off = l * (4 * l * l - 1) / 3;
    for (int ni = 0; ni < w; ni++)
      for (int ki = 0; ki < w; ki++)
        T[T_DEQ3 + off + ni * w + ki] = (float)wd(l, ni - l, ki - l, M_PI * 0.5);
  }
  for (int p = 0; p < 20; p++)
    for (int l = 0; l < 10; l++) {
      int w = 2 * l + 1, off = l * (4 * l * l - 1) / 3;
      for (int mi = 0; mi < w; mi++)
        for (int ni = 0; ni < w; ni++)
          T[T_DOUTF + p * 1330 + off + mi * w + ni] =
              (float)((2 * l + 1) * wd(l, mi - l, ni - l, bout[p]));
    }
  for (int p = 0; p < 20; p++) T[T_WOUT + p] = (float)wout[p];
}

extern "C" void kernel_launch(void* const* d_in, const int* in_sizes, int n_in,
                              void* d_out, int out_size, void* d_ws, size_t ws_size,
                              hipStream_t stream) {
  const float* x = (const float*)d_in[0];
  const float* w_s2 = (const float*)d_in[1];
  const float* b_s2 = (const float*)d_in[2];
  const float* bn1g = (const float*)d_in[3];
  const float* bn1b = (const float*)d_in[4];
  const float* w_so3 = (const float*)d_in[5];
  const float* b_so3 = (const float*)d_in[6];
  const float* bn2g = (const float*)d_in[7];
  const float* bn2b = (const float*)d_in[8];
  float* W = (float*)d_ws;
  float* out = (float*)d_out;

  // deterministic: tables rebuilt every call into a stable host buffer
  static std::vector<float> htab;
  htab.resize(TAB_N);
  build_tables(htab.data());
  (void)hipMemcpyAsync(W, htab.data(), TAB_N * sizeof(float),
                       hipMemcpyHostToDevice, stream);

  auto G1 = [](int n) { return dim3((unsigned)((n + 255) / 256)); };
  k_zero<<<G1(28800), 256, 0, stream>>>(W + O_IMG, 28800);
  k_project<<<G1(16384), 256, 0, stream>>>(x, W + O_IMG);
  k_adft<<<G1(14880), 256, 0, stream>>>(W);
  k_leg<<<G1(2048), 256, 0, stream>>>(W);
  k_wfs2<<<G1(3100), 256, 0, stream>>>(w_s2, W);
  k_s2conv<<<G1(4364800), 256, 0, stream>>>(W);
  k_so3_ifft1<<<25600, 128, 0, stream>>>(W, b_s2);
  k_bn_stats<<<100, 256, 0, stream>>>(W + O_Z, W + O_BN1, 32768);
  k_bn_apply<<<G1(26214400), 256, 0, stream>>>(W + O_Z, W + O_BN1, bn1g, bn1b,
                                               32768, 26214400);
  k_so3_fft<<<25600, 128, 0, stream>>>(W);
  k_wigf<<<G1(1064000), 256, 0, stream>>>(W);
  k_wft<<<G1(190000), 256, 0, stream>>>(w_so3, W);
  for (int l = 0; l < 10; l++) {
    int w = 2 * l + 1, M = 8 * w * w, N = 100 * w;
    k_at<<<G1(M * 100), 256, 0, stream>>>(W, l);
    dim3 gg((unsigned)((M + 31) / 32), (unsigned)((N + 63) / 64));
    k_cgemm<<<gg, 256, 0, stream>>>(W + O_AT, W + O_AT + 288800, W + O_WFT,
                                    W + O_WFT + 190000, W + O_G, W + O_G + 5487200,
                                    M, N, 100, 100, 1900, N, (9 - l) * 100);
    k_kred<<<G1(800 * w * w), 256, 0, stream>>>(W, l);
  }
  k_so3_ifft2<<<16000, 128, 0, stream>>>(W, b_so3);
  k_bn_stats<<<100, 256, 0, stream>>>(W + O_Z2, W + O_BN2, 8000);
  k_bn_apply<<<G1(6400000), 256, 0, stream>>>(W + O_Z2, W + O_BN2, bn2g, bn2b,
                                              8000, 6400000);
  k_final<<<800, 256, 0, stream>>>(W, out);
  (void)in_sizes;
  (void)n_in;
  (void)out_size;
  (void)ws_size;
}